// PredatorModel_3221225472652
// MI455X (gfx1250) — compile-verified
//
#include <hip/hip_runtime.h>

// LSTM (I=4, H=4) scan over T=2048, B=8192, fused with H->1 sigmoid head.
// One wave32 handles 16 batch columns via V_WMMA_F32_16X16X4_F32:
//   D[gate, batch] = A[gate, k] * B[k, batch] + C
// chained twice per timestep:  gates = W_hh*h + (W_ih*x_t + b)
// The W_ih*x WMMA for t+1 is issued inside step t (independent of recurrence).

typedef __attribute__((ext_vector_type(2))) float v2f;
typedef __attribute__((ext_vector_type(8))) float v8f;

#define T_STEPS 2048
#define BATCH   8192

__device__ __forceinline__ float wtanh(float x) {
#if __has_builtin(__builtin_amdgcn_tanhf)
  return __builtin_amdgcn_tanhf(x);
#else
  float r;
  asm("v_tanh_f32 %0, %1" : "=v"(r) : "v"(x));
  return r;
#endif
}

// exchange lanes n <-> n+16 (wave32), ds_swizzle group-of-32, xor_mask=0x10
__device__ __forceinline__ float swapx16(float x) {
  return __int_as_float(__builtin_amdgcn_ds_swizzle(__float_as_int(x), 0x401f));
}

__global__ __launch_bounds__(256, 1) void lstm_wmma16x16x4(
    const float* __restrict__ x,     // (T, B, 4)
    const float* __restrict__ w_ih,  // (16, 4)  gate order i,f,g,o
    const float* __restrict__ w_hh,  // (16, 4)
    const float* __restrict__ b_ih,  // (16,)
    const float* __restrict__ b_hh,  // (16,)
    const float* __restrict__ w_lin, // (1, 4)
    const float* __restrict__ b_lin, // (1,)
    float* __restrict__ out)         // (T, B, 1)
{
  const int lane  = threadIdx.x & 31;
  const int m     = lane & 15;   // row-in-half / batch column
  const int khalf = lane >> 4;   // 0 -> K={0,1}, 1 -> K={2,3}
  const int wave  = (int)((blockIdx.x * blockDim.x + threadIdx.x) >> 5);
  const int b0    = wave * 16;   // 16 batch elements per wave
  const bool lo   = (lane < 16);

  // ---- A operands (weights), resident across the whole scan --------------
  v2f a_ih, a_hh;
  {
    const float* p = w_ih + m * 4 + khalf * 2;
    a_ih.x = p[0]; a_ih.y = p[1];
    const float* q = w_hh + m * 4 + khalf * 2;
    a_hh.x = q[0]; a_hh.y = q[1];
  }

  // ---- bias in D layout: vgpr v holds gate (khalf*8 + v) ------------------
  v8f cbias;
  #pragma unroll
  for (int v = 0; v < 8; ++v)
    cbias[v] = b_ih[khalf * 8 + v] + b_hh[khalf * 8 + v];

  const float wl0 = w_lin[0], wl1 = w_lin[1], wl2 = w_lin[2], wl3 = w_lin[3];
  const float bl  = b_lin[0];

  // Activation constants for D slots 0..3:
  //   lanes 0-15 hold i-gates (sigmoid), lanes 16-31 hold g-gates (tanh).
  // sigmoid(z) = 0.5*tanh(0.5*z) + 0.5
  const float s0 = lo ? 0.5f : 1.0f;   // pre-scale
  const float a0 = lo ? 0.5f : 1.0f;   // post-scale
  const float c0 = lo ? 0.5f : 0.0f;   // post-offset
  // slots 4..7 (f-gates / o-gates) are sigmoid in both halves.

  // ---- streaming x: lane layout matches WMMA B operand --------------------
  // B (4x16): v0 = {K0 | K2}, v1 = {K1 | K3} across lane halves.
  const float* xp = x + (size_t)(b0 + m) * 4 + (size_t)khalf * 2;
  const size_t xstride = (size_t)BATCH * 4;

  v2f xv;
  xv.x = __builtin_nontemporal_load(xp + 0);
  xv.y = __builtin_nontemporal_load(xp + 1);
  const float* xq = xp + xstride;      // points at t+1

  // recurrent state (valid in lanes 0-15; high half carries bounded junk)
  float cs0 = 0.f, cs1 = 0.f, cs2 = 0.f, cs3 = 0.f;
  v2f bh; bh.x = 0.f; bh.y = 0.f;      // h operand, zero initial state

  // pipelined input projection for t=0
  v8f xg = __builtin_amdgcn_wmma_f32_16x16x4_f32(false, a_ih, false, xv,
                                                 (short)0, cbias, false, false);

  float* outp = out + b0;

  #pragma unroll 1
  for (int t = 0; t < T_STEPS; ++t) {
    // -- prefetch next x and its (recurrence-independent) projection --------
    v2f xnext = xv;
    if (t + 1 < T_STEPS) {
      xnext.x = __builtin_nontemporal_load(xq + 0);
      xnext.y = __builtin_nontemporal_load(xq + 1);
      xq += xstride;
    }

    // gates = W_hh * h + (W_ih * x_t + b)        (recurrent, critical path)
    v8f d = __builtin_amdgcn_wmma_f32_16x16x4_f32(false, a_hh, false, bh,
                                                  (short)0, xg, false, false);
    // input projection for t+1 (overlaps with activation chain below)
    v8f xgn = __builtin_amdgcn_wmma_f32_16x16x4_f32(false, a_ih, false, xnext,
                                                    (short)0, cbias, false, false);

    // -- activations: low half -> sigmoid(i), high half -> tanh(g) ----------
    float act0 = fmaf(wtanh(d[0] * s0), a0, c0);
    float act1 = fmaf(wtanh(d[1] * s0), a0, c0);
    float act2 = fmaf(wtanh(d[2] * s0), a0, c0);
    float act3 = fmaf(wtanh(d[3] * s0), a0, c0);
    // slots 4..7: sigmoid(f) in low half, sigmoid(o) in high half
    float act4 = fmaf(wtanh(d[4] * 0.5f), 0.5f, 0.5f);
    float act5 = fmaf(wtanh(d[5] * 0.5f), 0.5f, 0.5f);
    float act6 = fmaf(wtanh(d[6] * 0.5f), 0.5f, 0.5f);
    float act7 = fmaf(wtanh(d[7] * 0.5f), 0.5f, 0.5f);

    // -- bring tanh(g) and sigmoid(o) to the state-holding low half ---------
    float tg0 = swapx16(act0), tg1 = swapx16(act1);
    float tg2 = swapx16(act2), tg3 = swapx16(act3);
    float so0 = swapx16(act4), so1 = swapx16(act5);
    float so2 = swapx16(act6), so3 = swapx16(act7);

    // c = f*c + i*g   (low half: act0..3 = sigmoid(i), act4..7 = sigmoid(f))
    cs0 = fmaf(act4, cs0, act0 * tg0);
    cs1 = fmaf(act5, cs1, act1 * tg1);
    cs2 = fmaf(act6, cs2, act2 * tg2);
    cs3 = fmaf(act7, cs3, act3 * tg3);

    // h = o * tanh(c)
    float h0 = so0 * wtanh(cs0);
    float h1 = so1 * wtanh(cs1);
    float h2 = so2 * wtanh(cs2);
    float h3 = so3 * wtanh(cs3);

    // head: sigmoid(h . w_lin + b_lin)
    float y = fmaf(h0, wl0, fmaf(h1, wl1, fmaf(h2, wl2, fmaf(h3, wl3, bl))));
    y = fmaf(wtanh(y * 0.5f), 0.5f, 0.5f);
    if (lo)
      __builtin_nontemporal_store(y, outp + (size_t)t * BATCH + lane);

    // rebuild WMMA B operand of h: lanes<16 need {h0,h1}, lanes>=16 {h2,h3}
    float sh2 = swapx16(h2);   // lane n+16 receives h2 of batch n
    float sh3 = swapx16(h3);
    bh.x = lo ? h0 : sh2;
    bh.y = lo ? h1 : sh3;

    xv = xnext;
    xg = xgn;
  }
}

extern "C" void kernel_launch(void* const* d_in, const int* in_sizes, int n_in,
                              void* d_out, int out_size, void* d_ws, size_t ws_size,
                              hipStream_t stream) {
  (void)in_sizes; (void)n_in; (void)out_size; (void)d_ws; (void)ws_size;
  const float* x     = (const float*)d_in[0];
  const float* w_ih  = (const float*)d_in[1];
  const float* w_hh  = (const float*)d_in[2];
  const float* b_ih  = (const float*)d_in[3];
  const float* b_hh  = (const float*)d_in[4];
  const float* w_lin = (const float*)d_in[5];
  const float* b_lin = (const float*)d_in[6];
  float* out = (float*)d_out;

  // 8192 batch / 16 per wave = 512 waves; 8 waves (256 threads) per block
  dim3 block(256);
  dim3 grid((BATCH / 16) * 32 / 256);
  hipLaunchKernelGGL(lstm_wmma16x16x4, grid, block, 0, stream,
                     x, w_ih, w_hh, b_ih, b_hh, w_lin, b_lin, out);
}